// ContrastiveCosineLoss_21569325760642
// MI455X (gfx1250) — compile-verified
//
#include <hip/hip_runtime.h>
#include <hip/hip_bf16.h>

typedef __attribute__((ext_vector_type(2))) float v2f;
typedef __attribute__((ext_vector_type(8))) float v8f;

#define N 16384
#define D 64
#define NCLS 9
#define LDSTRIDE 68               // 68*4 = 272 bytes: 16B-aligned rows, conflict-free frag reads
#define FIXSCALE 16777216.0f      // 2^24 fixed point for deterministic integer accumulation

// ---------------- Kernel 1: softmax/valid/class + L2 normalize ----------------
__global__ void prep_kernel(const int* __restrict__ labels,
                            const float* __restrict__ feats,
                            const float* __restrict__ scores,
                            float* __restrict__ fn,
                            int* __restrict__ cls) {
    const int wave = threadIdx.x >> 5;
    const int lane = threadIdx.x & 31;
    const int row  = blockIdx.x * 8 + wave;   // grid = N/8 blocks of 256 = 8 waves

    // L2 normalize: each lane owns 2 of 64 dims
    const float2 x2 = reinterpret_cast<const float2*>(feats + (size_t)row * D)[lane];
    float ss = x2.x * x2.x + x2.y * x2.y;
    #pragma unroll
    for (int off = 16; off > 0; off >>= 1) ss += __shfl_xor(ss, off, 32);
    const float inv = 1.0f / fmaxf(sqrtf(ss), 1e-12f);
    float2 o; o.x = x2.x * inv; o.y = x2.y * inv;
    reinterpret_cast<float2*>(fn + (size_t)row * D)[lane] = o;

    if (lane == 0) {
        const float* s = scores + (size_t)row * NCLS;
        float m = s[0]; int am = 0;
        #pragma unroll
        for (int j = 1; j < NCLS; ++j) { float v = s[j]; if (v > m) { m = v; am = j; } }
        float se = 0.0f;
        #pragma unroll
        for (int j = 0; j < NCLS; ++j) se += expf(s[j] - m);
        const float pmax = 1.0f / se;                  // softmax prob of argmax
        const int lab = labels[row];
        const bool valid = (lab == am) && (pmax >= 0.5f);
        cls[row] = (valid && lab >= 1) ? lab : 0;      // 0 = excluded from loss
    }
}

// ---------------- Kernel 2: zero the 64 class-pair bins ----------------
__global__ void zero_bins_kernel(unsigned long long* __restrict__ gBins) {
    gBins[threadIdx.x] = 0ull;
}

// ---------------- Kernel 3: per-class cnt, ||sum fn||^2 -> pull part ----------------
__global__ void class_stats_kernel(const float* __restrict__ fn,
                                   const int* __restrict__ cls,
                                   float* __restrict__ cnts,
                                   float* __restrict__ pullParts) {
    const int c = blockIdx.x + 1;             // classes 1..8
    __shared__ float part[4][64];
    __shared__ float cpart[4];
    const int tid = threadIdx.x;
    const int d = tid & 63, g = tid >> 6;

    float acc = 0.0f, cn = 0.0f;
    for (int i = g; i < N; i += 4) {
        if (cls[i] == c) { acc += fn[(size_t)i * D + d]; cn += 1.0f; }
    }
    part[g][d] = acc;
    if (d == 0) cpart[g] = 0.0f;
    __syncthreads();
    if (d == 0) cpart[g] = cn;                 // cn identical across d, take d==0 copy
    __syncthreads();
    if (tid < 64) {
        float v = part[0][tid] + part[1][tid] + part[2][tid] + part[3][tid];
        part[0][tid] = v * v;
    }
    __syncthreads();
    if (tid == 0) {
        float quad = 0.0f;
        #pragma unroll
        for (int k = 0; k < 64; ++k) quad += part[0][k];   // = ||v_c||^2 = sum_{i,j in c} S_ij
        const float cnt = cpart[0] + cpart[1] + cpart[2] + cpart[3];
        cnts[blockIdx.x] = cnt;
        const float npairs  = cnt * (cnt - 1.0f) * 0.5f;
        const float pairSum = (quad - cnt) * 0.5f;
        pullParts[blockIdx.x] = (npairs > 0.0f) ? (1.0f - pairSum / npairs) : 0.0f;
    }
}

// ---------------- Kernel 4: tiled fp32 WMMA S = fn*fn^T, relu, class-pair binning ----------------
__global__ void pair_tiles_kernel(const float* __restrict__ fn,
                                  const int* __restrict__ cls,
                                  unsigned long long* __restrict__ gBins) {
    __shared__ float As[64 * LDSTRIDE];
    __shared__ float Bs[64 * LDSTRIDE];
    __shared__ int rcls[64];
    __shared__ int ccls[64];
    __shared__ unsigned long long bins[64];

    const int tid = threadIdx.x;
    const int rowBase = blockIdx.y * 64;
    const int colBase = blockIdx.x * 64;

    if (tid < 64) { bins[tid] = 0ull; rcls[tid] = cls[rowBase + tid]; }
    else if (tid < 128) { ccls[tid - 64] = cls[colBase + (tid - 64)]; }

    // Stage 64x64 fp32 panels into LDS (float4, 16B-aligned with LDSTRIDE=68)
    #pragma unroll
    for (int q = 0; q < 4; ++q) {
        const int f  = tid + q * 256;      // float4 index 0..1023
        const int r  = f >> 4;
        const int kc = (f & 15) << 2;
        *reinterpret_cast<float4*>(&As[r * LDSTRIDE + kc]) =
            *reinterpret_cast<const float4*>(fn + (size_t)(rowBase + r) * D + kc);
        *reinterpret_cast<float4*>(&Bs[r * LDSTRIDE + kc]) =
            *reinterpret_cast<const float4*>(fn + (size_t)(colBase + r) * D + kc);
    }
    __syncthreads();

    const int wave = tid >> 5;
    const int lane = tid & 31;
    const int half = lane >> 4;            // K-half for A/B frags, M-half for C/D
    const int mr   = lane & 15;            // row-of-A / col-of-B index within frag

    #pragma unroll
    for (int s = 0; s < 2; ++s) {
        const int st = wave + s * 8;       // 16 subtiles of 16x16, 8 waves x 2
        const int si = st >> 2;
        const int sj = st & 3;
        v8f acc = {};
        const float* ap = &As[(si * 16 + mr) * LDSTRIDE + half * 2];
        const float* bp = &Bs[(sj * 16 + mr) * LDSTRIDE + half * 2];
        #pragma unroll
        for (int t = 0; t < 16; ++t) {     // K = 64 in steps of 4
            v2f a = *reinterpret_cast<const v2f*>(ap + 4 * t);
            v2f b = *reinterpret_cast<const v2f*>(bp + 4 * t);
            acc = __builtin_amdgcn_wmma_f32_16x16x4_f32(
                false, a, false, b, (short)0, acc, false, false);
        }
        // C/D layout: lane -> col (lane&15), VGPR r -> row r + 8*(lane>>4)
        const int cc = ccls[sj * 16 + mr];
        if (cc > 0) {
            #pragma unroll
            for (int r = 0; r < 8; ++r) {
                const int cr = rcls[si * 16 + r + half * 8];
                if (cr > 0) {
                    const float v = acc[r] > 0.0f ? acc[r] : 0.0f;   // relu (MARGIN=0)
                    const unsigned long long q =
                        (unsigned long long)(v * FIXSCALE + 0.5f);
                    atomicAdd(&bins[(cr - 1) * 8 + (cc - 1)], q);    // ds_add_u64
                }
            }
        }
    }
    __syncthreads();
    if (tid < 64) atomicAdd(&gBins[tid], bins[tid]);                  // global u64 atomic
}

// ---------------- Kernel 5: combine pull + push ----------------
__global__ void finalize_kernel(const unsigned long long* __restrict__ gBins,
                                const float* __restrict__ cnts,
                                const float* __restrict__ pullParts,
                                float* __restrict__ out) {
    if (threadIdx.x == 0) {
        float pull = 0.0f;
        #pragma unroll
        for (int c = 0; c < 8; ++c) pull += pullParts[c];
        float push = 0.0f;
        for (int a = 0; a < 8; ++a) {
            for (int b = a + 1; b < 8; ++b) {
                const float denom = cnts[a] * cnts[b];
                if (denom > 0.0f) {
                    const double cross = (double)gBins[a * 8 + b] * (1.0 / 16777216.0);
                    push += (float)(cross / (double)denom);
                }
            }
        }
        out[0] = pull + push;
    }
}

extern "C" void kernel_launch(void* const* d_in, const int* in_sizes, int n_in,
                              void* d_out, int out_size, void* d_ws, size_t ws_size,
                              hipStream_t stream) {
    const int*   labels = (const int*)d_in[0];
    const float* feats  = (const float*)d_in[1];   // [1, N, D] flat
    const float* scores = (const float*)d_in[2];   // [N, 9]
    float* out = (float*)d_out;

    char* ws = (char*)d_ws;
    float* fn  = (float*)ws;                                       // N*D floats = 4 MB
    int*   cls = (int*)(ws + (size_t)N * D * sizeof(float));       // N ints
    unsigned long long* gBins =
        (unsigned long long*)(ws + (size_t)N * D * sizeof(float) + (size_t)N * sizeof(int));
    float* cnts      = (float*)((char*)gBins + 64 * sizeof(unsigned long long));
    float* pullParts = cnts + 8;

    prep_kernel<<<N / 8, 256, 0, stream>>>(labels, feats, scores, fn, cls);
    zero_bins_kernel<<<1, 64, 0, stream>>>(gBins);
    class_stats_kernel<<<8, 256, 0, stream>>>(fn, cls, cnts, pullParts);
    pair_tiles_kernel<<<dim3(N / 64, N / 64), 256, 0, stream>>>(fn, cls, gBins);
    finalize_kernel<<<1, 32, 0, stream>>>(gBins, cnts, pullParts, out);
}